// SymmetricTransitionUpBlock_9242769621756
// MI455X (gfx1250) — compile-verified
//
#include <hip/hip_runtime.h>
#include <math.h>

#define N1   60000
#define N2   15000
#define KNN  16
#define CIN  128
#define COUT 64
#define BN_EPS 1e-5f

typedef float v2f __attribute__((ext_vector_type(2)));
typedef float v8f __attribute__((ext_vector_type(8)));

// ---------------------------------------------------------------------------
// Kernel 0: init segment-softmax accumulators (must run every call: graph
// replays do not re-poison, and we need -inf / 0 starting state).
// ---------------------------------------------------------------------------
__global__ void init_seg_kernel(float* __restrict__ segmax, float* __restrict__ denom) {
    int i = blockIdx.x * blockDim.x + threadIdx.x;
    if (i < N1) {
        segmax[i] = -INFINITY;
        denom[i]  = 0.0f;
    }
}

// ---------------------------------------------------------------------------
// Kernel 1: fused WMMA GEMM over x2:
//   cols   0..127 : base    = x2 @ cs_w1[3:,:]          (raw, BN deferred)
//   cols 128..191 : up_feat = relu(bn2(x2 @ lin2_w))
// One 16x16 output tile per wave; K accumulated with v_wmma_f32_16x16x4_f32.
// B-source selection hoisted OUT of the K-loop (ptr/ldb/col picked once) so
// the inner loop is branch-free: no per-iteration EXEC save/restore.
// Layouts per CDNA5 ISA 7.12.2:
//   A 16x4 : lane M = l15, v = {A[M][k0+2*half], A[M][k0+2*half+1]}
//   B 4x16 : lane N = l15, v = {B[k0+2*half][N], B[k0+2*half+1][N]}
//   C 16x16: vgpr v -> M = v + 8*half, N = l15
// ---------------------------------------------------------------------------
__global__ void gemm_x2_kernel(const float* __restrict__ x2,
                               const float* __restrict__ cs_w1,   // (131,128)
                               const float* __restrict__ lin2_w,  // (128,64)
                               const float* __restrict__ bn2_g, const float* __restrict__ bn2_b,
                               const float* __restrict__ bn2_m, const float* __restrict__ bn2_v,
                               float* __restrict__ base,          // (N2,128)
                               float* __restrict__ up_feat)       // (N2,64)
{
    const int NT = 12;                 // 8 base tiles + 4 up_feat tiles
    const int MT = (N2 + 15) / 16;     // 938
    int wave = blockIdx.x * (blockDim.x >> 5) + (threadIdx.x >> 5);
    if (wave >= MT * NT) return;       // wave-uniform: EXEC stays all-ones for WMMA
    int mt = wave / NT;
    int nt = wave % NT;
    int lane = threadIdx.x & 31;
    int half = lane >> 4;
    int l15  = lane & 15;
    int r0   = mt * 16;
    int arow = r0 + l15; if (arow >= N2) arow = N2 - 1;   // clamp; stores guarded

    // Hoisted B selection: branch once, inner loop unconditional.
    const float* Bp;
    int ldb, col;
    if (nt < 8) {
        Bp  = cs_w1 + 3 * CIN;   // rows 3..130 of cs_w1
        ldb = CIN;
        col = nt * 16 + l15;
    } else {
        Bp  = lin2_w;
        ldb = COUT;
        col = (nt - 8) * 16 + l15;
    }
    const float* Arow = x2 + (size_t)arow * CIN;
    const float* Bcol = Bp + col + (size_t)(half * 2) * ldb;  // row k0+2*half, this col
    const int    ldb2 = 4 * ldb;                              // advance 4 K-rows / step

    v8f acc = {};
    #pragma unroll 4
    for (int k0 = 0; k0 < CIN; k0 += 4) {
        int kk = k0 + half * 2;
        v2f a; a.x = Arow[kk]; a.y = Arow[kk + 1];
        v2f b; b.x = Bcol[0];  b.y = Bcol[ldb];
        Bcol += ldb2;
        acc = __builtin_amdgcn_wmma_f32_16x16x4_f32(false, a, false, b,
                                                    (short)0, acc, false, false);
    }

    if (nt < 8) {
        #pragma unroll
        for (int v = 0; v < 8; ++v) {
            int row = r0 + v + half * 8;
            if (row < N2) base[(size_t)row * CIN + col] = acc[v];
        }
    } else {
        float s  = rsqrtf(bn2_v[col] + BN_EPS) * bn2_g[col];
        float mm = bn2_m[col], bb = bn2_b[col];
        #pragma unroll
        for (int v = 0; v < 8; ++v) {
            int row = r0 + v + half * 8;
            float h = (acc[v] - mm) * s + bb;
            h = h > 0.0f ? h : 0.0f;
            if (row < N2) up_feat[(size_t)row * COUT + col] = h;
        }
    }
}

// ---------------------------------------------------------------------------
// Kernel 2: per-(m,k) logit. One wave per neighbor pair (240000 waves).
// val[c] = base[m][c] + pr . cs_w1[0:3,c];  h = relu(bn_cs(val));
// logit = h . cs_w2 + b2 via wave32 shuffle reduction; atomicMax into segmax.
// ---------------------------------------------------------------------------
__global__ void logits_kernel(const float* __restrict__ p1, const float* __restrict__ p2,
                              const int*   __restrict__ knn,
                              const float* __restrict__ base,
                              const float* __restrict__ cs_w1,
                              const float* __restrict__ gcs, const float* __restrict__ bcs,
                              const float* __restrict__ mcs, const float* __restrict__ vcs,
                              const float* __restrict__ w2,  const float* __restrict__ b2,
                              float* __restrict__ logits, float* __restrict__ segmax)
{
    int wave = blockIdx.x * (blockDim.x >> 5) + (threadIdx.x >> 5); // exactly 240000
    int lane = threadIdx.x & 31;
    int m   = wave >> 4;
    int idx = knn[wave];
    float prx = p1[idx * 3 + 0] - p2[m * 3 + 0];
    float pry = p1[idx * 3 + 1] - p2[m * 3 + 1];
    float prz = p1[idx * 3 + 2] - p2[m * 3 + 2];

    int c0 = lane * 4;                       // 32 lanes x 4 = 128 channels
    float acc = 0.0f;
    #pragma unroll
    for (int j = 0; j < 4; ++j) {
        int c = c0 + j;
        float val = base[(size_t)m * CIN + c]
                  + prx * cs_w1[c]
                  + pry * cs_w1[CIN + c]
                  + prz * cs_w1[2 * CIN + c];
        float h = (val - mcs[c]) * rsqrtf(vcs[c] + BN_EPS) * gcs[c] + bcs[c];
        h = h > 0.0f ? h : 0.0f;
        acc += h * w2[c];
    }
    #pragma unroll
    for (int off = 16; off > 0; off >>= 1)
        acc += __shfl_xor(acc, off, 32);

    if (lane == 0) {
        float l = acc + b2[0];
        logits[wave] = l;
        atomicMax(segmax + idx, l);
    }
}

// ---------------------------------------------------------------------------
// Kernel 3: ex = exp(logit - segmax[flat]); denom[flat] += ex (in-place over
// the logits buffer).
// ---------------------------------------------------------------------------
__global__ void exp_denom_kernel(const int* __restrict__ knn,
                                 const float* __restrict__ segmax,
                                 float* __restrict__ logits_ex,
                                 float* __restrict__ denom)
{
    int i = blockIdx.x * blockDim.x + threadIdx.x;
    if (i >= N2 * KNN) return;
    int idx = knn[i];
    float e = expf(logits_ex[i] - segmax[idx]);
    logits_ex[i] = e;
    atomicAdd(denom + idx, e);
}

// ---------------------------------------------------------------------------
// Kernel 4: y_base = relu(bn1(x1 @ lin1_w)) -> d_out (scatter adds on top).
// Same WMMA tiling; M=60000 (3750 tiles), N=64 (4 tiles), K=64.
// ---------------------------------------------------------------------------
__global__ void gemm_y_kernel(const float* __restrict__ x1,
                              const float* __restrict__ lin1_w,   // (64,64)
                              const float* __restrict__ bn1_g, const float* __restrict__ bn1_b,
                              const float* __restrict__ bn1_m, const float* __restrict__ bn1_v,
                              float* __restrict__ out)            // (N1,64)
{
    const int NT = 4;
    const int MT = N1 / 16;            // 3750 exact
    int wave = blockIdx.x * (blockDim.x >> 5) + (threadIdx.x >> 5);
    if (wave >= MT * NT) return;
    int mt = wave / NT;
    int nt = wave % NT;
    int lane = threadIdx.x & 31;
    int half = lane >> 4;
    int l15  = lane & 15;
    int r0   = mt * 16;
    int col  = nt * 16 + l15;

    const float* Arow = x1 + (size_t)(r0 + l15) * COUT;
    const float* Bcol = lin1_w + col + (size_t)(half * 2) * COUT;

    v8f acc = {};
    #pragma unroll 4
    for (int k0 = 0; k0 < COUT; k0 += 4) {
        int kk = k0 + half * 2;
        v2f a; a.x = Arow[kk]; a.y = Arow[kk + 1];
        v2f b; b.x = Bcol[0];  b.y = Bcol[COUT];
        Bcol += 4 * COUT;
        acc = __builtin_amdgcn_wmma_f32_16x16x4_f32(false, a, false, b,
                                                    (short)0, acc, false, false);
    }
    float s  = rsqrtf(bn1_v[col] + BN_EPS) * bn1_g[col];
    float mm = bn1_m[col], bb = bn1_b[col];
    #pragma unroll
    for (int v = 0; v < 8; ++v) {
        int row = r0 + v + half * 8;
        float h = (acc[v] - mm) * s + bb;
        h = h > 0.0f ? h : 0.0f;
        out[(size_t)row * COUT + col] = h;
    }
}

// ---------------------------------------------------------------------------
// Kernel 5: out[flat[i]] += (ex[i]/denom[flat[i]]) * up_feat[m].  4 channels
// per thread; lanes of a half-wave share one row -> broadcast loads.
// ---------------------------------------------------------------------------
__global__ void scatter_kernel(const int* __restrict__ knn,
                               const float* __restrict__ ex,
                               const float* __restrict__ denom,
                               const float* __restrict__ up_feat,
                               float* __restrict__ out)
{
    int gid = blockIdx.x * blockDim.x + threadIdx.x;   // N2*KNN*16 threads
    if (gid >= N2 * KNN * 16) return;
    int i  = gid >> 4;           // neighbor-pair row
    int c0 = (gid & 15) * 4;     // channel group
    int idx = knn[i];
    int m   = i >> 4;
    float p = ex[i] / denom[idx];
    const float* uf = up_feat + (size_t)m * COUT + c0;
    float*       o  = out     + (size_t)idx * COUT + c0;
    atomicAdd(o + 0, uf[0] * p);
    atomicAdd(o + 1, uf[1] * p);
    atomicAdd(o + 2, uf[2] * p);
    atomicAdd(o + 3, uf[3] * p);
}

// ---------------------------------------------------------------------------
extern "C" void kernel_launch(void* const* d_in, const int* in_sizes, int n_in,
                              void* d_out, int out_size, void* d_ws, size_t ws_size,
                              hipStream_t stream) {
    const float* p1     = (const float*)d_in[0];
    const float* x1     = (const float*)d_in[1];
    const float* p2     = (const float*)d_in[2];
    const float* x2     = (const float*)d_in[3];
    const int*   knn    = (const int*)  d_in[4];
    const float* lin1_w = (const float*)d_in[5];
    const float* bn1_g  = (const float*)d_in[6];
    const float* bn1_b  = (const float*)d_in[7];
    const float* bn1_m  = (const float*)d_in[8];
    const float* bn1_v  = (const float*)d_in[9];
    const float* lin2_w = (const float*)d_in[10];
    const float* bn2_g  = (const float*)d_in[11];
    const float* bn2_b  = (const float*)d_in[12];
    const float* bn2_m  = (const float*)d_in[13];
    const float* bn2_v  = (const float*)d_in[14];
    const float* cs_w1  = (const float*)d_in[15];
    const float* bncs_g = (const float*)d_in[16];
    const float* bncs_b = (const float*)d_in[17];
    const float* bncs_m = (const float*)d_in[18];
    const float* bncs_v = (const float*)d_in[19];
    const float* cs_w2  = (const float*)d_in[20];
    const float* cs_b2  = (const float*)d_in[21];
    float* out = (float*)d_out;

    // workspace layout (floats), total ~13 MB
    float* wsf     = (float*)d_ws;
    float* base    = wsf;                          // N2*CIN   = 1,920,000
    float* up_feat = base    + (size_t)N2 * CIN;   // N2*COUT  =   960,000
    float* ex      = up_feat + (size_t)N2 * COUT;  // N2*KNN   =   240,000
    float* segmax  = ex      + (size_t)N2 * KNN;   // N1       =    60,000
    float* denom   = segmax  + N1;                 // N1       =    60,000
    (void)ws_size; (void)in_sizes; (void)n_in; (void)out_size;

    // 0) init softmax accumulators
    init_seg_kernel<<<(N1 + 255) / 256, 256, 0, stream>>>(segmax, denom);

    // 1) fused x2 GEMM (base + up_feat):  938 M-tiles * 12 N-tiles = 11256 waves
    {
        int waves = ((N2 + 15) / 16) * 12;
        int blocks = (waves + 7) / 8;
        gemm_x2_kernel<<<blocks, 256, 0, stream>>>(x2, cs_w1, lin2_w,
                                                   bn2_g, bn2_b, bn2_m, bn2_v,
                                                   base, up_feat);
    }

    // 2) logits + segment max: 240000 waves = 30000 blocks of 8 waves
    logits_kernel<<<(N2 * KNN) / 8, 256, 0, stream>>>(p1, p2, knn, base, cs_w1,
                                                      bncs_g, bncs_b, bncs_m, bncs_v,
                                                      cs_w2, cs_b2, ex, segmax);

    // 3) exp + segment sum
    exp_denom_kernel<<<(N2 * KNN + 255) / 256, 256, 0, stream>>>(knn, segmax, ex, denom);

    // 4) y = relu(bn1(x1 @ lin1_w)) -> out (3750*4 = 15000 waves)
    gemm_y_kernel<<<(N1 / 16 * 4 + 7) / 8, 256, 0, stream>>>(x1, lin1_w,
                                                             bn1_g, bn1_b, bn1_m, bn1_v,
                                                             out);

    // 5) scatter prob-weighted up_feat into out
    scatter_kernel<<<(N2 * KNN * 16 + 255) / 256, 256, 0, stream>>>(knn, ex, denom,
                                                                    up_feat, out);
}